// SSD_ResNet_75453985456329
// MI455X (gfx1250) — compile-verified
//
#include <hip/hip_runtime.h>
#include <hip/hip_bf16.h>
#include <math.h>

// ---------------------------------------------------------------------------
// SSD-1D detector on MI455X (gfx1250).
// Conv FLOPs on V_WMMA_F32_16X16X4_F32 (full fp32 matrix pipe, matches ref).
// Per-tap GEMM decomposition (K = Cin): no division in the hot loop,
// wave-uniform interior test per (tile, tap). Unsigned 32-bit strength-
// reduced offsets -> loads use SGPR-base + zext 32-bit VGPR-offset
// (global_load_b32 v, voff, s[base] scale_offset), no 64-bit address math.
// ---------------------------------------------------------------------------

typedef float v2f __attribute__((ext_vector_type(2)));
typedef float v8f __attribute__((ext_vector_type(8)));

#define NUM_PRIORS 3024
#define TOPK 200

// ---------------------------------------------------------------------------
// Fast conv1d + bias + optional ReLU. Requires Cin % 32 == 0.
// Grid: x = ceil(xTiles/8), y = coTiles, z = batch. Block = 256 (8 wave32).
// WMMA operand mapping (CDNA5 ISA 7.12.2, 32-bit A 16x4 / B 4x16 / C 16x16):
//   A/B: K = kstep + vgprIdx + 2*(lane>=16); C/D: VGPR r -> rows {r,r+8}, N=lane&15
// ---------------------------------------------------------------------------
__global__ void conv1d_wmma_fast(
    const float* __restrict__ in, const float* __restrict__ wgt,
    const float* __restrict__ bias, float* __restrict__ out,
    int Cin, int inPitch, int Lin,
    int Cout, int outPitch, int Lout,
    int ksz, int stride, int pad, int doRelu)
{
    const int lane = threadIdx.x & 31;
    const int wave = threadIdx.x >> 5;
    const int xTiles = (Lout + 15) >> 4;
    const int xT = blockIdx.x * (blockDim.x >> 5) + wave;
    if (xT >= xTiles) return;                       // wave-uniform exit

    const int coT = blockIdx.y;
    const int b   = blockIdx.z;

    const int half   = lane >> 4;                   // K/row MSB select
    const int lo     = lane & 15;
    const int xbase  = xT * 16;
    const int cobase = coT * 16;
    const int n      = xbase + lo;                  // output position (this lane)
    const int m      = cobase + lo;                 // weight row (this lane)

    const int  mC    = (m < Cout) ? m : (Cout - 1);
    const bool mFull = (cobase + 15) < Cout;        // wave-uniform
    const bool mOK   = (m < Cout);

    // unsigned 32-bit element offsets (all buffers < 2^31 elements, all
    // clamped offsets non-negative) -> zext addressing, no sign-extension.
    const unsigned inBase = (unsigned)(b * inPitch * Lin);   // wave-uniform
    const unsigned wBase  = (unsigned)(mC * Cin * ksz);
    const unsigned wStep  = (unsigned)(4 * ksz);             // K+4 advance
    const unsigned iStep  = (unsigned)(4 * Lin);

    v8f acc = {0.f,0.f,0.f,0.f,0.f,0.f,0.f,0.f};

    for (int t = 0; t < ksz; ++t) {
        const int pos = n * stride - pad + t;
        // wave-uniform interior test for this tap
        const bool safe = (xbase * stride - pad + t >= 0) &&
                          ((xbase + 15) * stride - pad + t < Lin) &&
                          ((xbase + 15) < Lout) && mFull;
        unsigned wOff0 = wBase + (unsigned)((2 * half) * ksz + t);
        unsigned wOff1 = wOff0 + (unsigned)ksz;
        if (safe) {
            unsigned iOff0 = inBase + (unsigned)((2 * half) * Lin + pos);
            unsigned iOff1 = iOff0 + (unsigned)Lin;
            for (int k0 = 0; k0 < Cin; k0 += 32) {
                #pragma unroll
                for (int u = 0; u < 8; ++u) {
                    v2f a, bv;
                    a.x  = wgt[wOff0];
                    a.y  = wgt[wOff1];
                    bv.x = in [iOff0];
                    bv.y = in [iOff1];
                    wOff0 += wStep; wOff1 += wStep;
                    iOff0 += iStep; iOff1 += iStep;
                    acc = __builtin_amdgcn_wmma_f32_16x16x4_f32(
                              false, a, false, bv, (short)0, acc, false, false);
                }
            }
        } else {
            // clamp addresses (always in-bounds) + cndmask selects: no exec
            // divergence, loads issued unconditionally.
            const int  posC = pos < 0 ? 0 : (pos >= Lin ? Lin - 1 : pos);
            const bool pOK  = (pos >= 0) && (pos < Lin) && (n < Lout);
            unsigned iOff0 = inBase + (unsigned)((2 * half) * Lin + posC);
            unsigned iOff1 = iOff0 + (unsigned)Lin;
            for (int k0 = 0; k0 < Cin; k0 += 32) {
                #pragma unroll
                for (int u = 0; u < 8; ++u) {
                    v2f a, bv;
                    const float ax = wgt[wOff0];
                    const float ay = wgt[wOff1];
                    const float b0 = in [iOff0];
                    const float b1 = in [iOff1];
                    wOff0 += wStep; wOff1 += wStep;
                    iOff0 += iStep; iOff1 += iStep;
                    a.x  = mOK ? ax : 0.0f;
                    a.y  = mOK ? ay : 0.0f;
                    bv.x = pOK ? b0 : 0.0f;
                    bv.y = pOK ? b1 : 0.0f;
                    acc = __builtin_amdgcn_wmma_f32_16x16x4_f32(
                              false, a, false, bv, (short)0, acc, false, false);
                }
            }
        }
    }

    if (n < Lout) {
        const unsigned outBase = (unsigned)(b * outPitch * Lout);
        #pragma unroll
        for (int r = 0; r < 8; ++r) {
            const int mm = cobase + r + 8 * half;
            if (mm < Cout) {
                float v = acc[r] + bias[mm];
                if (doRelu) v = fmaxf(v, 0.0f);
                out[outBase + (unsigned)(mm * Lout + n)] = v;
            }
        }
    }
}

// ---------------------------------------------------------------------------
// Generic conv (used only for layer 0: Cin=1, ksz=7). Flattened K = Cin*ksz
// stepped by 4, clamp+select guards (no divergence).
// ---------------------------------------------------------------------------
__global__ void conv1d_wmma_generic(
    const float* __restrict__ in, const float* __restrict__ wgt,
    const float* __restrict__ bias, float* __restrict__ out,
    int Cin, int inPitch, int Lin,
    int Cout, int outPitch, int Lout,
    int ksz, int stride, int pad, int doRelu)
{
    const int lane = threadIdx.x & 31;
    const int wave = threadIdx.x >> 5;
    const int xTiles = (Lout + 15) >> 4;
    const int xT = blockIdx.x * (blockDim.x >> 5) + wave;
    if (xT >= xTiles) return;
    const int coT = blockIdx.y;
    const int b   = blockIdx.z;

    const int half = lane >> 4;
    const int lo   = lane & 15;
    const int n    = xT * 16 + lo;
    const int m    = coT * 16 + lo;
    const int Ktot = Cin * ksz;

    const int      mC     = (m < Cout) ? m : (Cout - 1);
    const unsigned wBase  = (unsigned)(mC * Ktot);
    const bool     mOK    = (m < Cout);
    const unsigned inBase = (unsigned)(b * inPitch * Lin);
    const bool     nOK    = (n < Lout);

    v8f acc = {0.f,0.f,0.f,0.f,0.f,0.f,0.f,0.f};

    for (int k0 = 0; k0 < Ktot; k0 += 4) {
        const int ka = k0 + 2 * half;
        v2f a, bv;
        {
            const int ka0 = ka     < Ktot ? ka     : (Ktot - 1);
            const int ka1 = ka + 1 < Ktot ? ka + 1 : (Ktot - 1);
            const float a0 = wgt[wBase + (unsigned)ka0];
            const float a1 = wgt[wBase + (unsigned)ka1];
            a.x = (mOK && ka     < Ktot) ? a0 : 0.0f;
            a.y = (mOK && ka + 1 < Ktot) ? a1 : 0.0f;
        }
        #pragma unroll
        for (int j = 0; j < 2; ++j) {
            const int kk  = ka + j;
            const int kkC = kk < Ktot ? kk : (Ktot - 1);
            const int ci  = kkC / ksz;
            const int t   = kkC - ci * ksz;
            const int pos = n * stride - pad + t;
            const int posC = pos < 0 ? 0 : (pos >= Lin ? Lin - 1 : pos);
            const float v = in[inBase + (unsigned)(ci * Lin + posC)];
            const bool ok = (kk < Ktot) && nOK && (pos >= 0) && (pos < Lin);
            if (j == 0) bv.x = ok ? v : 0.0f; else bv.y = ok ? v : 0.0f;
        }
        acc = __builtin_amdgcn_wmma_f32_16x16x4_f32(
                  false, a, false, bv, (short)0, acc, false, false);
    }

    if (nOK) {
        const unsigned outBase = (unsigned)(b * outPitch * Lout);
        #pragma unroll
        for (int r = 0; r < 8; ++r) {
            const int mm = coT * 16 + r + 8 * half;
            if (mm < Cout) {
                float v = acc[r] + bias[mm];
                if (doRelu) v = fmaxf(v, 0.0f);
                out[outBase + (unsigned)(mm * Lout + n)] = v;
            }
        }
    }
}

// ---------------------------------------------------------------------------
// Decode: priors inline, softmax over 7 classes, box decode.
// ---------------------------------------------------------------------------
__global__ void decode_softmax(const float* __restrict__ loc,
                               const float* __restrict__ conf,
                               float* __restrict__ boxes,
                               float* __restrict__ scores, int nB)
{
    const int gid = blockIdx.x * blockDim.x + threadIdx.x;
    if (gid >= nB * NUM_PRIORS) return;
    const int b = gid / NUM_PRIORS;
    const int p = gid - b * NUM_PRIORS;

    const int flen[6]  = {512, 256, 128, 64, 32, 16};
    const int pref[6]  = {0, 1536, 2304, 2688, 2880, 2976};
    const int fpref[6] = {0, 512, 768, 896, 960, 992};

    int j = 0;
    while (j < 5 && p >= pref[j + 1]) ++j;
    const int r = p - pref[j];
    const int x = r / 3;
    const int q = r - 3 * x;
    const int f = flen[j];

    const float* lj = loc  + (size_t)6  * nB * fpref[j];
    const float* cj = conf + (size_t)21 * nB * fpref[j];

    float pcx = ((float)x + 0.5f) / (float)f;
    float pw  = (float)(q + 1) / (float)f;
    pcx = fminf(fmaxf(pcx, 0.0f), 1.0f);
    pw  = fminf(fmaxf(pw,  0.0f), 1.0f);

    const float l0 = lj[((size_t)b * 6 + (q * 2 + 0)) * f + x];
    const float l1 = lj[((size_t)b * 6 + (q * 2 + 1)) * f + x];
    const float cx = pcx + l0 * 0.1f * pw;
    const float w  = pw * expf(l1 * 0.2f);
    boxes[((size_t)b * NUM_PRIORS + p) * 2 + 0] = cx - 0.5f * w;
    boxes[((size_t)b * NUM_PRIORS + p) * 2 + 1] = cx + 0.5f * w;

    float lg[7];
    float mx = -3.402823466e+38f;
    #pragma unroll
    for (int c = 0; c < 7; ++c) {
        lg[c] = cj[((size_t)b * 21 + (q * 7 + c)) * f + x];
        mx = fmaxf(mx, lg[c]);
    }
    float sum = 0.0f;
    #pragma unroll
    for (int c = 0; c < 7; ++c) { lg[c] = expf(lg[c] - mx); sum += lg[c]; }
    const float inv = 1.0f / sum;
    #pragma unroll
    for (int c = 1; c < 7; ++c)
        scores[((size_t)b * 6 + (c - 1)) * NUM_PRIORS + p] = lg[c] * inv;
}

// ---------------------------------------------------------------------------
// Per-(batch,class): threshold -> top-200 -> greedy NMS -> rows [score,x1,x2].
// ---------------------------------------------------------------------------
__global__ void topk_nms(const float* __restrict__ scores,
                         const float* __restrict__ boxes,
                         float* __restrict__ out)
{
    __shared__ float ssc[NUM_PRIORS];
    __shared__ float rv[256];
    __shared__ int   ri[256];
    __shared__ float tx1[TOPK], tx2[TOPK], tsc[TOPK];
    __shared__ int   keep[TOPK];

    const int b   = blockIdx.x / 6;
    const int cls = blockIdx.x - b * 6;
    const int tid = threadIdx.x;

    const float* sp = scores + ((size_t)b * 6 + cls) * NUM_PRIORS;
    for (int i = tid; i < NUM_PRIORS; i += blockDim.x) {
        float v = sp[i];
        ssc[i] = (v > 0.01f) ? v : 0.0f;           // CONF_THRESH
    }
    __syncthreads();

    for (int sel = 0; sel < TOPK; ++sel) {
        float bv = -1.0f; int bi = 0x7fffffff;
        for (int i = tid; i < NUM_PRIORS; i += blockDim.x) {
            float v = ssc[i];
            if (v > bv || (v == bv && i < bi)) { bv = v; bi = i; }
        }
        rv[tid] = bv; ri[tid] = bi;
        __syncthreads();
        for (int off = 128; off > 0; off >>= 1) {
            if (tid < off) {
                float v2 = rv[tid + off]; int i2 = ri[tid + off];
                if (v2 > rv[tid] || (v2 == rv[tid] && i2 < ri[tid])) {
                    rv[tid] = v2; ri[tid] = i2;
                }
            }
            __syncthreads();
        }
        if (tid == 0) {
            const int   idx = ri[0];
            const float v   = rv[0];
            tsc[sel]  = v;
            tx1[sel]  = boxes[((size_t)b * NUM_PRIORS + idx) * 2 + 0];
            tx2[sel]  = boxes[((size_t)b * NUM_PRIORS + idx) * 2 + 1];
            keep[sel] = (v > 0.0f) ? 1 : 0;
            ssc[idx]  = -1.0f;
        }
        __syncthreads();
    }

    if (tid == 0) {
        for (int i = 1; i < TOPK; ++i) {
            if (!keep[i]) continue;
            const float xi1 = tx1[i], xi2 = tx2[i];
            const float ai  = xi2 - xi1;
            for (int j = 0; j < i; ++j) {
                if (!keep[j]) continue;
                float inter = fminf(tx2[j], xi2) - fmaxf(tx1[j], xi1);
                inter = fmaxf(inter, 0.0f);
                const float aj  = tx2[j] - tx1[j];
                const float iou = inter / (aj + ai - inter + 1e-9f);
                if (iou > 0.1f) { keep[i] = 0; break; }   // NMS_THRESH
            }
        }
    }
    __syncthreads();

    float* op = out + ((size_t)b * 7 + (cls + 1)) * TOPK * 3;
    for (int i = tid; i < TOPK; i += blockDim.x) {
        if (keep[i]) {
            op[i * 3 + 0] = tsc[i];
            op[i * 3 + 1] = tx1[i];
            op[i * 3 + 2] = tx2[i];
        } else {
            op[i * 3 + 0] = 0.0f;
            op[i * 3 + 1] = 0.0f;
            op[i * 3 + 2] = 0.0f;
        }
    }
}

__global__ void zero_f32(float* __restrict__ p, int n)
{
    const int i = blockIdx.x * blockDim.x + threadIdx.x;
    if (i < n) p[i] = 0.0f;
}

// ---------------------------------------------------------------------------
// Host orchestration
// ---------------------------------------------------------------------------
static void launch_conv(hipStream_t st, const float* in, const float* w,
                        const float* bias, float* out,
                        int nB, int Cin, int inPitch, int Lin,
                        int Cout, int outPitch, int Lout,
                        int k, int s, int p, int relu)
{
    const int coTiles = (Cout + 15) / 16;
    const int xTiles  = (Lout + 15) / 16;
    const int wavesPerBlock = 8;                 // 256 threads = 8 wave32
    dim3 grid((xTiles + wavesPerBlock - 1) / wavesPerBlock, coTiles, nB);
    if ((Cin & 31) == 0) {
        conv1d_wmma_fast<<<grid, 256, 0, st>>>(in, w, bias, out,
            Cin, inPitch, Lin, Cout, outPitch, Lout, k, s, p, relu);
    } else {
        conv1d_wmma_generic<<<grid, 256, 0, st>>>(in, w, bias, out,
            Cin, inPitch, Lin, Cout, outPitch, Lout, k, s, p, relu);
    }
}

extern "C" void kernel_launch(void* const* d_in, const int* in_sizes, int n_in,
                              void* d_out, int out_size, void* d_ws, size_t ws_size,
                              hipStream_t stream)
{
    (void)in_sizes; (void)n_in; (void)out_size; (void)ws_size;

    static const int SPEC[12][5] = {
        {1, 64, 7, 2, 3}, {64, 64, 3, 2, 1}, {64, 64, 3, 1, 1},
        {64, 128, 3, 2, 1}, {128, 128, 3, 1, 1}, {128, 128, 3, 2, 1},
        {128, 128, 3, 1, 1}, {128, 128, 3, 2, 1}, {128, 128, 3, 2, 1},
        {128, 128, 3, 2, 1}, {128, 128, 3, 2, 1}, {128, 128, 3, 2, 1}};
    static const int FLEN[6]  = {512, 256, 128, 64, 32, 16};
    static const int FPREF[6] = {0, 512, 768, 896, 960, 992};
    const int B = 32;

    // inputs in setup_inputs() dict (insertion) order
    const float* x[3] = {(const float*)d_in[0], (const float*)d_in[1],
                         (const float*)d_in[2]};
    const float *bw[3][12], *bb[3][12];
    int idx = 3;
    for (int br = 0; br < 3; ++br)
        for (int l = 0; l < 12; ++l) {
            bw[br][l] = (const float*)d_in[idx++];
            bb[br][l] = (const float*)d_in[idx++];
        }
    const float *locw[6], *locb[6], *confw[6], *confb[6];
    for (int j = 0; j < 6; ++j) { locw[j]  = (const float*)d_in[idx++];
                                  locb[j]  = (const float*)d_in[idx++]; }
    for (int j = 0; j < 6; ++j) { confw[j] = (const float*)d_in[idx++];
                                  confb[j] = (const float*)d_in[idx++]; }

    // workspace layout (floats)
    float* ws = (float*)d_ws;
    float* bufA   = ws;                                  // 8,388,608
    float* bufB   = bufA + (size_t)8388608;              // 4,194,304
    float* srcAll = bufB + (size_t)4194304;              // 12,386,304
    float* locAll = srcAll + (size_t)12386304;           // 193,536
    float* cfAll  = locAll + (size_t)193536;             // 677,376
    float* boxes  = cfAll  + (size_t)677376;             // 193,536
    float* scores = boxes  + (size_t)193536;             // 580,608

    // backbone: 3 branches through 12 conv+relu layers
    for (int br = 0; br < 3; ++br) {
        const float* cur = x[br];
        int curPitch = 1, Lin = 8192;
        for (int l = 0; l < 12; ++l) {
            const int Cin = SPEC[l][0], Cout = SPEC[l][1], k = SPEC[l][2];
            const int s = SPEC[l][3], p = SPEC[l][4];
            const int Lout = (Lin + 2 * p - k) / s + 1;
            float* dst; int outPitch;
            if (l < 6) {
                dst = (l & 1) ? bufB : bufA;
                outPitch = Cout;
            } else {
                const int j = l - 6;
                dst = srcAll + (size_t)384 * B * FPREF[j] + (size_t)br * 128 * Lout;
                outPitch = 384;
            }
            launch_conv(stream, cur, bw[br][l], bb[br][l], dst,
                        B, Cin, curPitch, Lin, Cout, outPitch, Lout, k, s, p, 1);
            cur = dst; curPitch = outPitch; Lin = Lout;
        }
    }

    // detection heads
    for (int j = 0; j < 6; ++j) {
        const int f = FLEN[j];
        const float* src = srcAll + (size_t)384 * B * FPREF[j];
        float* loP = locAll + (size_t)6  * B * FPREF[j];
        float* cfP = cfAll  + (size_t)21 * B * FPREF[j];
        launch_conv(stream, src, locw[j],  locb[j],  loP, B, 384, 384, f,
                    6, 6, f, 3, 1, 1, 0);
        launch_conv(stream, src, confw[j], confb[j], cfP, B, 384, 384, f,
                    21, 21, f, 3, 1, 1, 0);
    }

    // decode + softmax
    {
        const int total = B * NUM_PRIORS;
        decode_softmax<<<(total + 255) / 256, 256, 0, stream>>>(
            locAll, cfAll, boxes, scores, B);
    }

    // output: zero (background stays 0), then per-(b,cls) topk+NMS
    {
        const int n = B * 7 * TOPK * 3;
        zero_f32<<<(n + 255) / 256, 256, 0, stream>>>((float*)d_out, n);
        topk_nms<<<B * 6, 256, 0, stream>>>(scores, boxes, (float*)d_out);
    }
}